// WaveletDenoiseLayer_49778670960864
// MI455X (gfx1250) — compile-verified
//
#include <hip/hip_runtime.h>
#include <math.h>

typedef __attribute__((ext_vector_type(2))) float v2f;
typedef __attribute__((ext_vector_type(8))) float v8f;

#define B_  64
#define T_  4096
#define C_  128

// db4 decomposition filters (pywt), UNFLIPPED (used directly for reconstruction
// correlation; dwt uses flipped indexing [7-j]).
__constant__ float c_lo[8] = {
    -0.010597401784997278f,  0.032883011666982945f,  0.030841381835986965f,
    -0.18703481171888114f,  -0.02798376941698385f,   0.6308807679295904f,
     0.7148465705525415f,    0.23037781330885523f };
__constant__ float c_hi[8] = {
    -0.23037781330885523f,   0.7148465705525415f,   -0.6308807679295904f,
    -0.02798376941698385f,   0.18703481171888114f,   0.030841381835986965f,
    -0.032883011666982945f, -0.010597401784997278f };

// ---------------------------------------------------------------------------
// Tiled transposes: (B,T,C) <-> row-major (B*C, T)
// ---------------------------------------------------------------------------
__global__ __launch_bounds__(256) void transpose_in_kernel(
    const float* __restrict__ x, float* __restrict__ xs)
{
    __shared__ float tile[32][33];
    const int b  = blockIdx.z;
    const int c0 = blockIdx.x * 32;
    const int t0 = blockIdx.y * 32;
    const int tx = threadIdx.x, ty = threadIdx.y;
#pragma unroll
    for (int i = 0; i < 32; i += 8)
        tile[ty + i][tx] = x[(size_t)b * T_ * C_ + (size_t)(t0 + ty + i) * C_ + (c0 + tx)];
    __syncthreads();
#pragma unroll
    for (int i = 0; i < 32; i += 8)
        xs[((size_t)b * C_ + (c0 + ty + i)) * T_ + (t0 + tx)] = tile[tx][ty + i];
}

__global__ __launch_bounds__(256) void transpose_out_kernel(
    const float* __restrict__ ys, float* __restrict__ out)
{
    __shared__ float tile[32][33];
    const int b  = blockIdx.z;
    const int c0 = blockIdx.x * 32;
    const int t0 = blockIdx.y * 32;
    const int tx = threadIdx.x, ty = threadIdx.y;
#pragma unroll
    for (int i = 0; i < 32; i += 8)
        tile[ty + i][tx] = ys[((size_t)b * C_ + (c0 + ty + i)) * T_ + (t0 + tx)];
    __syncthreads();
#pragma unroll
    for (int i = 0; i < 32; i += 8)
        out[(size_t)b * T_ * C_ + (size_t)(t0 + ty + i) * C_ + (c0 + tx)] = tile[tx][ty + i];
}

// ---------------------------------------------------------------------------
// Analysis filterbank via V_WMMA_F32_16X16X4_F32.
//   out[o]   = sum_j ext[2o+j] * filt[7-j]       (stride-2 valid correlation)
//   D = A(16x4) x B(4x16) + C :  A[M][K] = ext[2(o0+M)+jbase+K]
//   B col0 = flipped lo, col1 = flipped hi  ->  D[M][0]=ca, D[M][1]=cd
// A layout (32-bit 16x4): lanes0-15 M rows, VGPR0->K0,VGPR1->K1; lanes16-31 K2/K3.
// B layout (32-bit 4x16): lane&15 = N; VGPR0: K0 (lanes<16) / K2; VGPR1: K1 / K3.
// D layout (16x16 f32):   VGPR r, lanes0-15 -> M=r, lanes16-31 -> M=8+r; N=lane&15.
// ca/cd buffers are padded to a full 16-tile so the 8-element result store is
// unconditional (vectorizes to ds_store_b128 x2; no exec-mask churn).
// ---------------------------------------------------------------------------
__device__ __forceinline__ void dwt_wmma(const float* ext, float* ca, float* cd,
                                         int ntiles,
                                         int lane, int wave, int nwaves)
{
    const int n  = lane & 15;
    const int hi = lane >> 4;

    const int jx0 = hi ? 2 : 0;   // VGPR0 K index for jbase=0
    const int jy0 = hi ? 3 : 1;   // VGPR1 K index for jbase=0
    const int jx1 = jx0 + 4, jy1 = jy0 + 4;

    v2f b0, b1;
    b0.x = (n == 0) ? c_lo[7 - jx0] : ((n == 1) ? c_hi[7 - jx0] : 0.0f);
    b0.y = (n == 0) ? c_lo[7 - jy0] : ((n == 1) ? c_hi[7 - jy0] : 0.0f);
    b1.x = (n == 0) ? c_lo[7 - jx1] : ((n == 1) ? c_hi[7 - jx1] : 0.0f);
    b1.y = (n == 0) ? c_lo[7 - jy1] : ((n == 1) ? c_hi[7 - jy1] : 0.0f);

    const int m  = lane & 15;
    const int k0 = hi ? 2 : 0;

    for (int t = wave; t < ntiles; t += nwaves) {
        const int o0 = t << 4;
        const float* base = ext + 2 * (o0 + m) + k0;
        v2f a0;  a0.x  = base[0]; a0.y  = base[1];
        v2f a1v; a1v.x = base[4]; a1v.y = base[5];

        v8f acc = {0.f, 0.f, 0.f, 0.f, 0.f, 0.f, 0.f, 0.f};
        acc = __builtin_amdgcn_wmma_f32_16x16x4_f32(false, a0,  false, b0,
                                                    (short)0, acc, false, false);
        acc = __builtin_amdgcn_wmma_f32_16x16x4_f32(false, a1v, false, b1,
                                                    (short)0, acc, false, false);
        if (n < 2) {
            float* dst = ((n == 0) ? ca : cd) + o0 + hi * 8;
#pragma unroll
            for (int r = 0; r < 8; ++r)
                dst[r] = acc[r];   // unconditional: buffers padded to tile edge
        }
    }
}

// ---------------------------------------------------------------------------
// Fused per-row denoise: one workgroup (256 thr = 8 waves) per signal row.
//   ext(sym) -> DWT1 (WMMA) -> DWT2 (WMMA) -> median radix-select -> soft thr
//   -> IDWT2 -> IDWT1 -> store
// LDS footprint ~57.9 KB.
// ---------------------------------------------------------------------------
__global__ __launch_bounds__(256) void denoise_kernel(
    const float* __restrict__ in, int inMode,
    float* __restrict__ outp,     int outMode)
{
    __shared__ float s_ext [4136];  // L=4096: ext len 4109, padded for wmma tail
    __shared__ float s_ca1 [2064];  // 2051 valid, padded to 129 tiles * 16
    __shared__ float s_cd1 [2064];
    __shared__ float s_ext2[2088];  // L=2051: ext len 2064, padded
    __shared__ float s_ca2 [1040];  // 1029 valid, padded to 65 tiles * 16
    __shared__ float s_cd2 [1040];
    __shared__ float s_a1  [2052];
    __shared__ int   s_cnt;

    const int tid  = threadIdx.x;
    const int lane = tid & 31;
    const int wave = tid >> 5;
    const int r    = blockIdx.x;          // row in (B*C, T)

    size_t ibase, obase; int istr, ostr;
    if (inMode)  { ibase = (size_t)(r >> 7) * T_ * C_ + (r & 127); istr = C_; }
    else         { ibase = (size_t)r * T_;                         istr = 1;  }
    if (outMode) { obase = (size_t)(r >> 7) * T_ * C_ + (r & 127); ostr = C_; }
    else         { obase = (size_t)r * T_;                         ostr = 1;  }

    // ---- level-1 symmetric extension (pad 7/7, drop first -> 6 left, 7 right)
    for (int p = tid; p < 4136; p += 256) {
        float v = 0.0f;
        if (p < 4109) {
            int i   = p - 6;
            int idx = (i < 0) ? (-i - 1) : ((i < T_) ? i : (2 * T_ - 1 - i));
            v = in[ibase + (size_t)idx * istr];
        }
        s_ext[p] = v;
    }
    __syncthreads();

    // ---- level-1 DWT: 2051 valid outputs, 129 tiles of 16
    dwt_wmma(s_ext, s_ca1, s_cd1, 129, lane, wave, 8);
    __syncthreads();

    // ---- level-2 extension of ca1 (L=2051)
    for (int p = tid; p < 2088; p += 256) {
        float v = 0.0f;
        if (p < 2064) {
            int i   = p - 6;
            int idx = (i < 0) ? (-i - 1) : ((i < 2051) ? i : (2 * 2051 - 1 - i));
            v = s_ca1[idx];
        }
        s_ext2[p] = v;
    }
    __syncthreads();

    // ---- level-2 DWT: 1029 valid outputs, 65 tiles
    dwt_wmma(s_ext2, s_ca2, s_cd2, 65, lane, wave, 8);
    __syncthreads();

    // ---- exact median of |cd1| (2051 elems, 0-based index 1025 -> k=1026)
    // radix/binary select on the u32 pattern (nonneg floats are order-isomorphic)
    unsigned lo = 0u, hiv = 0xffffffffu;
    for (int it = 0; it < 32 && lo < hiv; ++it) {
        unsigned mid = lo + ((hiv - lo) >> 1);
        if (tid == 0) s_cnt = 0;
        __syncthreads();
        int local = 0;
        for (int i = tid; i < 2051; i += 256) {
            unsigned u = __float_as_uint(s_cd1[i]) & 0x7fffffffu;
            local += (u <= mid) ? 1 : 0;
        }
        atomicAdd(&s_cnt, local);
        __syncthreads();
        const int tot = s_cnt;
        __syncthreads();
        if (tot >= 1026) hiv = mid; else lo = mid + 1;
    }
    const float med = __uint_as_float(lo);
    const float thr = (med / 0.6745f) * sqrtf(2.0f * logf((float)T_));

    // ---- soft threshold cd2 and cd1 in place
    for (int i = tid; i < 1029; i += 256) {
        float c = s_cd2[i];
        float a = fabsf(c) - thr;
        s_cd2[i] = (a > 0.0f) ? copysignf(a, c) : 0.0f;
    }
    for (int i = tid; i < 2051; i += 256) {
        float c = s_cd1[i];
        float a = fabsf(c) - thr;
        s_cd1[i] = (a > 0.0f) ? copysignf(a, c) : 0.0f;
    }
    __syncthreads();

    // ---- IDWT level 2 -> a1 (trim to 2051)
    // out[t] = sum_j up[t+j-1]*K[j], up[2i]=ca (lo), up[2i+1]=cd (hi)
    for (int t = tid; t < 2051; t += 256) {
        float acc = 0.0f;
#pragma unroll
        for (int j = 0; j < 8; ++j) {
            int p = t + j - 1;
            if ((unsigned)p < 2058u) {  // 2*1029
                if (p & 1) acc += s_cd2[p >> 1] * c_hi[j];
                else       acc += s_ca2[p >> 1] * c_lo[j];
            }
        }
        s_a1[t] = acc;
    }
    __syncthreads();

    // ---- IDWT level 1 -> write 4096 outputs straight to global
    for (int t = tid; t < T_; t += 256) {
        float acc = 0.0f;
#pragma unroll
        for (int j = 0; j < 8; ++j) {
            int p = t + j - 1;
            if ((unsigned)p < 4102u) {  // 2*2051
                if (p & 1) acc += s_cd1[p >> 1] * c_hi[j];
                else       acc += s_a1[p >> 1]  * c_lo[j];
            }
        }
        outp[obase + (size_t)t * ostr] = acc;
    }
}

// ---------------------------------------------------------------------------
extern "C" void kernel_launch(void* const* d_in, const int* in_sizes, int n_in,
                              void* d_out, int out_size, void* d_ws, size_t ws_size,
                              hipStream_t stream)
{
    (void)in_sizes; (void)n_in; (void)out_size;
    const float* x   = (const float*)d_in[0];
    float*       out = (float*)d_out;

    const size_t plane = (size_t)B_ * T_ * C_;
    const size_t bytes = plane * sizeof(float);

    float* xs = (float*)d_ws;
    float* ys = xs + plane;
    const bool haveXs = (ws_size >= bytes);
    const bool haveYs = (ws_size >= 2 * bytes);

    dim3 gt(C_ / 32, T_ / 32, B_);   // (4, 128, 64)
    dim3 bt(32, 8, 1);

    if (haveXs)
        transpose_in_kernel<<<gt, bt, 0, stream>>>(x, xs);

    const float* din  = haveXs ? (const float*)xs : x;
    float*       dout = haveYs ? ys : out;
    denoise_kernel<<<B_ * C_, 256, 0, stream>>>(din, haveXs ? 0 : 1,
                                                dout, haveYs ? 0 : 1);

    if (haveYs)
        transpose_out_kernel<<<gt, bt, 0, stream>>>(ys, out);
}